// SkeletonExtraction_28106265985415
// MI455X (gfx1250) — compile-verified
//
#include <hip/hip_runtime.h>

// Skeleton extraction: Otsu binarize -> Zhang-Suen thinning (to convergence) -> 3x3 dilation.
// One workgroup (256 threads, 8 wave32) per 256x256 image; 512 images.
// Single HBM read + single HBM write (~268 MB total -> ~11.5us floor @ 23.3 TB/s);
// all intermediate state lives in LDS.
// CDNA5 paths: global_load_async_to_lds_b128 / global_store_async_from_lds_b128 +
// s_wait_asynccnt, wave32 ballot, bit-sliced thinning (32 px per u32 word).

#define IMG_H 256
#define IMG_W 256
#define NPIX  (IMG_H * IMG_W)   // 65536
#define NTHR  256
#define WPR   8                  // u32 words per bit-packed row

__device__ __forceinline__ void async_load_b128(unsigned int lds_off, const void* gaddr) {
    asm volatile("global_load_async_to_lds_b128 %0, %1, off"
                 :: "v"(lds_off), "v"(gaddr) : "memory");
}
__device__ __forceinline__ void async_store_b128(const void* gaddr, unsigned int lds_off) {
    asm volatile("global_store_async_from_lds_b128 %0, %1, off"
                 :: "v"(gaddr), "v"(lds_off) : "memory");
}
__device__ __forceinline__ void wait_async0() {
#if __has_builtin(__builtin_amdgcn_s_wait_asynccnt)
    __builtin_amdgcn_s_wait_asynccnt(0);
#else
    asm volatile("s_wait_asynccnt 0x0" ::: "memory");
#endif
}
__device__ __forceinline__ unsigned int ballot32(bool pred) {
#if __has_builtin(__builtin_amdgcn_ballot_w32)
    return __builtin_amdgcn_ballot_w32(pred);
#else
    return (unsigned int)__ballot(pred);
#endif
}

// full adder on bit-planes: s + 2*cy = a + b + c
__device__ __forceinline__ void fadd(unsigned int a, unsigned int b, unsigned int c,
                                     unsigned int& s, unsigned int& cy) {
    unsigned int t = a ^ b;
    s  = t ^ c;
    cy = (a & b) | (c & t);
}
// bit-sliced popcount of 8 one-bit planes: result = o1 + 2*o2 + 4*o4 + 8*o8 per bit lane
__device__ __forceinline__ void add8(unsigned int n0, unsigned int n1, unsigned int n2,
                                     unsigned int n3, unsigned int n4, unsigned int n5,
                                     unsigned int n6, unsigned int n7,
                                     unsigned int& o1, unsigned int& o2,
                                     unsigned int& o4, unsigned int& o8) {
    unsigned int s1, c1, s2, c2, s3, c3, ca, sc, cc, cd;
    fadd(n0, n1, n2, s1, c1);
    fadd(n3, n4, n5, s2, c2);
    s3 = n6 ^ n7; c3 = n6 & n7;
    fadd(s1, s2, s3, o1, ca);      // weight-1 result, carry into weight-2
    fadd(c1, c2, c3, sc, cc);      // weight-2 partial, carry into weight-4
    o2 = sc ^ ca; cd = sc & ca;    // weight-2 final
    o4 = cc ^ cd; o8 = cc & cd;    // weight-4, weight-8
}

__global__ __launch_bounds__(NTHR) void skeleton_kernel(const float* __restrict__ in,
                                                        float* __restrict__ out) {
    extern __shared__ unsigned char smem[];
    unsigned char* imgA     = smem;                                   // 64 KB: byte values (phase 1-3)
    float*         stage    = (float*)(smem + 65536);                 // 64 KB: async-DMA staging
    unsigned int*  bitA     = (unsigned int*)(smem + 131072);         // 8 KB: bit image (ping)
    unsigned int*  bitB     = (unsigned int*)(smem + 139264);         // 8 KB: bit image (pong) / dilation
    unsigned int*  hist     = (unsigned int*)(smem + 147456);         // 1 KB
    float*         sigma_sh = (float*)(smem + 148480);                // 1 KB
    int*           thresh_p = (int*)(smem + 149504);
    unsigned int*  chg_p    = (unsigned int*)(smem + 149508);

    const int tid = threadIdx.x;
    const int img = blockIdx.x;
    const float* gsrc = in  + (size_t)img * NPIX;
    float*       gdst = out + (size_t)img * NPIX;

    hist[tid] = 0u;

    // ---------- Phase 1: async-DMA input (8 chunks x 8192 floats, double buffered),
    //            convert to byte values + LDS histogram ----------
    auto issue_load = [&](int c, int half) {
        const float* gb = gsrc + c * 8192;
        unsigned int lb = (unsigned int)(uintptr_t)(stage + half * 8192);
        #pragma unroll
        for (int j = 0; j < 8; ++j) {
            int e = j * 1024 + tid * 4;                 // 16 B per lane per issue
            async_load_b128(lb + (unsigned int)(e * 4), (const void*)(gb + e));
        }
    };
    issue_load(0, 0);
    for (int c = 0; c < 8; ++c) {
        wait_async0();            // this wave's portion of chunk c landed
        __syncthreads();          // all waves' portions landed (also covers hist zeroing)
        if (c + 1 < 8) issue_load(c + 1, (c + 1) & 1);
        const float* s = stage + (c & 1) * 8192;
        #pragma unroll 4
        for (int j = 0; j < 32; ++j) {
            int e = j * 256 + tid;
            float f = s[e];
            unsigned int v = ((unsigned int)(int)(f * 255.0f)) & 255u;  // uint8 truncation semantics
            imgA[c * 8192 + e] = (unsigned char)v;
            atomicAdd(&hist[v], 1u);
        }
        __syncthreads();
    }

    // ---------- Phase 2: Otsu threshold (thread t evaluates candidate threshold t) ----------
    {
        float w1 = 0.f, m1 = 0.f, muT = 0.f;
        for (int i = 0; i < 256; ++i) {
            float h  = (float)hist[i];
            float hi = h * (float)i;
            muT += hi;
            if (i <= tid) { w1 += h; m1 += hi; }
        }
        float w2  = (float)NPIX - w1;
        float num = m1 * w2 - (muT - m1) * w1;
        num = num * num;
        sigma_sh[tid] = (w1 > 0.f && w2 > 0.f) ? (num / fmaxf(w1 * w2, 1.0f)) : 0.f;
    }
    __syncthreads();
    if (tid == 0) {            // first-occurrence argmax (matches jnp.argmax)
        float best = sigma_sh[0]; int bi = 0;
        for (int i = 1; i < 256; ++i)
            if (sigma_sh[i] > best) { best = sigma_sh[i]; bi = i; }
        *thresh_p = bi;
    }
    __syncthreads();

    // ---------- Phase 3: binarize -> bit-packed image via wave32 ballot ----------
    {
        const int t   = *thresh_p;
        const int wv  = tid >> 5;          // wave id (0..7)
        const int ln  = tid & 31;          // lane id
        for (int k = 0; k < 256; ++k) {    // 8 waves x 256 words = 2048 words
            int w = k * 8 + wv;
            bool pred = (int)imgA[w * 32 + ln] > t;     // bit i == pixel x = 32w+i
            unsigned int m = ballot32(pred);
            if (ln == 0) bitA[w] = m;
        }
    }
    __syncthreads();

    // ---------- Phase 4: bit-sliced Zhang-Suen thinning to convergence ----------
    // thread y processes row y: 8 words, 32 pixels per word, sliding prev/cur/next words.
    auto substep = [&](const unsigned int* S, unsigned int* D, int step) {
        const int y = tid;
        const unsigned int* rT = (y > 0)         ? S + (y - 1) * WPR : nullptr;
        const unsigned int* rM =                   S +  y      * WPR;
        const unsigned int* rB = (y < IMG_H - 1) ? S + (y + 1) * WPR : nullptr;
        unsigned int* dR = D + y * WPR;
        unsigned int anyrm = 0u;
        unsigned int tP = 0, mP = 0, bP = 0;
        unsigned int tC = rT ? rT[0] : 0u, mC = rM[0], bC = rB ? rB[0] : 0u;
        #pragma unroll
        for (int w = 0; w < WPR; ++w) {
            unsigned int tN = 0, mN = 0, bN = 0;
            if (w < WPR - 1) {
                tN = rT ? rT[w + 1] : 0u;
                mN = rM[w + 1];
                bN = rB ? rB[w + 1] : 0u;
            }
            // neighbor bitboards; bit i = pixel x = 32w+i (LSB = lowest x)
            unsigned int n0 = tC;                          // N
            unsigned int n4 = bC;                          // S
            unsigned int n2 = (mC >> 1) | (mN << 31);      // E
            unsigned int n6 = (mC << 1) | (mP >> 31);      // W
            unsigned int n1 = (tC >> 1) | (tN << 31);      // NE
            unsigned int n7 = (tC << 1) | (tP >> 31);      // NW
            unsigned int n3 = (bC >> 1) | (bN << 31);      // SE
            unsigned int n5 = (bC << 1) | (bP >> 31);      // SW
            // B = neighbor count (bit-sliced)
            unsigned int b1, b2, b4, b8;
            add8(n0, n1, n2, n3, n4, n5, n6, n7, b1, b2, b4, b8);
            // A = 0->1 transitions around the ring (bit-sliced)
            unsigned int e0 = ~n0 & n1, e1 = ~n1 & n2, e2 = ~n2 & n3, e3 = ~n3 & n4;
            unsigned int e4 = ~n4 & n5, e5 = ~n5 & n6, e6 = ~n6 & n7, e7 = ~n7 & n0;
            unsigned int a1, a2, a4, a8;
            add8(e0, e1, e2, e3, e4, e5, e6, e7, a1, a2, a4, a8);
            unsigned int A1  = a1 & ~a2 & ~a4 & ~a8;                       // A == 1
            unsigned int okB = (b2 | b4) & ~b8 & ~(b1 & b2 & b4);          // 2 <= B <= 6
            unsigned int cc  = (step == 0)
                ? (~(n0 & n2 & n4) & ~(n2 & n4 & n6))     // P2*P4*P6==0 && P4*P6*P8==0
                : (~(n0 & n2 & n6) & ~(n0 & n4 & n6));    // P2*P4*P8==0 && P2*P6*P8==0
            unsigned int rm = mC & okB & A1 & cc;
            dR[w]  = mC & ~rm;
            anyrm |= rm;
            tP = tC; mP = mC; bP = bC;
            tC = tN; mC = mN; bC = bN;
        }
        if (anyrm) atomicOr(chg_p, 1u);
    };

    for (;;) {
        if (tid == 0) *chg_p = 0u;
        __syncthreads();
        substep(bitA, bitB, 0);
        __syncthreads();
        substep(bitB, bitA, 1);     // each full iteration lands back in bitA
        __syncthreads();
        unsigned int ch = *chg_p;
        __syncthreads();
        if (!ch) break;
    }

    // ---------- Phase 5: 3x3 max dilation on bitboards -> bitB ----------
    {
        const int y = tid;
        const unsigned int* rT = (y > 0)         ? bitA + (y - 1) * WPR : nullptr;
        const unsigned int* rM =                   bitA +  y      * WPR;
        const unsigned int* rB = (y < IMG_H - 1) ? bitA + (y + 1) * WPR : nullptr;
        auto or3 = [&](int w) {
            unsigned int v = rM[w];
            if (rT) v |= rT[w];
            if (rB) v |= rB[w];
            return v;
        };
        unsigned int rP = 0, rC = or3(0);
        #pragma unroll
        for (int w = 0; w < WPR; ++w) {
            unsigned int rN = (w < WPR - 1) ? or3(w + 1) : 0u;
            bitB[y * WPR + w] = rC | ((rC >> 1) | (rN << 31)) | ((rC << 1) | (rP >> 31));
            rP = rC; rC = rN;
        }
    }
    __syncthreads();

    // ---------- Phase 6: expand bits -> float, async-DMA LDS -> HBM (4 chunks x 64 rows) ----------
    for (int c4 = 0; c4 < 4; ++c4) {
        __syncthreads();                       // stage free (prev chunk's stores waited below)
        #pragma unroll 2
        for (int j = 0; j < 64; ++j) {
            int p = j * 256 + tid;             // 0..16383 within chunk
            int y = c4 * 64 + (p >> 8);
            int x = p & 255;
            unsigned int wd = bitB[y * WPR + (x >> 5)];
            stage[p] = (float)((wd >> (x & 31)) & 1u);
        }
        __syncthreads();
        const float* gb = gdst + c4 * 16384;
        unsigned int lb = (unsigned int)(uintptr_t)stage;
        #pragma unroll
        for (int j = 0; j < 16; ++j) {
            int e = j * 1024 + tid * 4;
            async_store_b128((const void*)(gb + e), lb + (unsigned int)(e * 4));
        }
        wait_async0();                         // drain before stage reuse / kernel end
    }
}

extern "C" void kernel_launch(void* const* d_in, const int* in_sizes, int n_in,
                              void* d_out, int out_size, void* d_ws, size_t ws_size,
                              hipStream_t stream) {
    (void)n_in; (void)out_size; (void)d_ws; (void)ws_size;
    const float* feat = (const float*)d_in[0];
    float* out = (float*)d_out;
    const int n_img = in_sizes[0] / NPIX;                  // (8*64*256*256)/65536 = 512
    const size_t smem_bytes = 149520;                      // ~146 KB (within 320 KB/WGP, 2 WGs/WGP)
    skeleton_kernel<<<dim3(n_img), dim3(NTHR), smem_bytes, stream>>>(feat, out);
}